// MAS_34823594836438
// MI455X (gfx1250) — compile-verified
//
#include <hip/hip_runtime.h>
#include <math.h>

// Problem constants (match reference)
#define F_INN   1433
#define HIDW    128
#define NHEAD   8
#define CHAN    16
#define NCLS    7
#define NEGSL   0.2f

typedef float v2f __attribute__((ext_vector_type(2)));
typedef float v8f __attribute__((ext_vector_type(8)));

static __device__ __forceinline__ float leakyf(float x) {
    return x > 0.f ? x : NEGSL * x;
}

// float atomic max via sign-split int/uint atomics (monotone toward max, race-safe)
static __device__ __forceinline__ void atomicMaxF(float* addr, float v) {
    if (v >= 0.f) atomicMax((int*)addr, __float_as_int(v));
    else          atomicMin((unsigned int*)addr, __float_as_uint(v));
}

// ---------------- degree / normalization ----------------
__global__ void k_init_deg(float* __restrict__ deg, int n) {
    int i = blockIdx.x * blockDim.x + threadIdx.x;
    if (i < n) deg[i] = 1.0f;                    // self-loop
}
__global__ void k_deg_edges(const int* __restrict__ dst, float* __restrict__ deg, int e) {
    int i = blockIdx.x * blockDim.x + threadIdx.x;
    if (i < e) atomicAdd(&deg[dst[i]], 1.0f);
}
__global__ void k_rsqrt(float* __restrict__ deg, int n) {
    int i = blockIdx.x * blockDim.x + threadIdx.x;
    if (i < n) deg[i] = rsqrtf(deg[i]);          // deg >= 1 always
}

// ---------------- weight transpose: Bt[n][k] = B[k][n], row stride Kp (zero padded) ----
// Makes each lane's WMMA B-fragment a contiguous 8B-aligned pair -> global_load_b64.
__global__ void k_transposeB(const float* __restrict__ B, float* __restrict__ Bt,
                             int K, int Kp) {
    long i = (long)blockIdx.x * blockDim.x + threadIdx.x;
    if (i >= (long)HIDW * Kp) return;
    int nIdx = (int)(i / Kp);
    int k    = (int)(i % Kp);
    Bt[i] = (k < K) ? B[(size_t)k * HIDW + nIdx] : 0.f;
}

// ---------------- WMMA f32 GEMM: C[M x 128] = A[M x K] @ B[K x 128] ----------------
// Block = 256 threads = 8 waves; block computes 16 rows x 128 cols.
// A 16-row strip staged in LDS (shared by all 8 waves); B read transposed (Bt) so the
// per-lane fragment pair is one b64 load. Main K-loop is fully unguarded; zero-padded
// Bt/LDS make the tail chunk branch-free inside the inner loop too.
#define KCHUNK 64
#define LDSS   66   // padded stride: bank = (2*row + k) % 64 -> conflict-free half-wave reads

__global__ __launch_bounds__(256)
void k_gemm_wmma128(const float* __restrict__ A, const float* __restrict__ Bt,
                    float* __restrict__ C, int M, int K, int Kp) {
    __shared__ float As[16 * LDSS];
    const int tid  = threadIdx.x;
    const int wave = tid >> 5;
    const int lane = tid & 31;
    const int half = lane >> 4;          // 0 or 1
    const int l16  = lane & 15;
    const int m0   = blockIdx.x * 16;
    const int n0   = wave * 16;

    // lane's column of Bt (row-contiguous in k)
    const float* __restrict__ Bl = Bt + (size_t)(n0 + l16) * Kp;
    const float* __restrict__ Al = As + l16 * LDSS + 2 * half;

    v8f acc = {0.f, 0.f, 0.f, 0.f, 0.f, 0.f, 0.f, 0.f};

    int k0 = 0;
    // ---- main loop: full 64-wide chunks, no per-step guards ----
    for (; k0 + KCHUNK <= K; k0 += KCHUNK) {
        __syncthreads();
#pragma unroll
        for (int r = 0; r < 4; ++r) {
            int idx = tid + 256 * r;                 // 0..1023 over 16x64 tile
            int row = idx >> 6;
            int kk  = idx & 63;
            float v = (m0 + row < M) ? A[(size_t)(m0 + row) * K + (k0 + kk)] : 0.f;
            As[row * LDSS + kk] = v;
        }
        __syncthreads();
#pragma unroll
        for (int s = 0; s < 16; ++s) {
            float2 av = *(const float2*)&Al[4 * s];           // ds_load_b64
            float2 bv = *(const float2*)&Bl[k0 + 4 * s + 2 * half]; // global_load_b64
            v2f a; a[0] = av.x; a[1] = av.y;
            v2f b; b[0] = bv.x; b[1] = bv.y;
            acc = __builtin_amdgcn_wmma_f32_16x16x4_f32(
                      false, a, false, b, (short)0, acc, false, false);
        }
    }
    // ---- tail chunk (< 64 wide): zero-filled LDS + zero-padded Bt, still branch-free inner ----
    if (k0 < K) {
        __syncthreads();
#pragma unroll
        for (int r = 0; r < 4; ++r) {
            int idx = tid + 256 * r;
            int row = idx >> 6;
            int kk  = idx & 63;
            float v = 0.f;
            if ((k0 + kk) < K && (m0 + row) < M)
                v = A[(size_t)(m0 + row) * K + (k0 + kk)];
            As[row * LDSS + kk] = v;
        }
        __syncthreads();
        int ksteps = (K - k0 + 3) >> 2;
        for (int s = 0; s < ksteps; ++s) {
            float2 av = *(const float2*)&Al[4 * s];
            float2 bv = *(const float2*)&Bl[k0 + 4 * s + 2 * half];
            v2f a; a[0] = av.x; a[1] = av.y;
            v2f b; b[0] = bv.x; b[1] = bv.y;
            acc = __builtin_amdgcn_wmma_f32_16x16x4_f32(
                      false, a, false, b, (short)0, acc, false, false);
        }
    }
    // C/D layout: VGPR r -> row = r + 8*(L/16), col = L%16
#pragma unroll
    for (int r = 0; r < 8; ++r) {
        int row = r + 8 * half;
        if (m0 + row < M)
            C[(size_t)(m0 + row) * HIDW + n0 + l16] = acc[r];
    }
}

// ---------------- GCN aggregation ----------------
// out[i][f] = hW[i][f]*dinv[i]^2 + bias[f]  (self-loop + bias init)
__global__ void k_gcn_init(const float* __restrict__ hW, const float* __restrict__ dinv,
                           const float* __restrict__ bias, float* __restrict__ out, int n) {
    long i = (long)blockIdx.x * blockDim.x + threadIdx.x;
    if (i < (long)n * HIDW) {
        int node = (int)(i >> 7), f = (int)(i & 127);
        float di = dinv[node];
        out[i] = hW[i] * di * di + bias[f];
    }
}
// wave-per-edge scatter: lane handles 4 consecutive features
__global__ __launch_bounds__(256)
void k_gcn_scatter(const float* __restrict__ hW, const float* __restrict__ dinv,
                   const int* __restrict__ src, const int* __restrict__ dst,
                   float* __restrict__ out, int e) {
    int wid  = (int)(((long)blockIdx.x * blockDim.x + threadIdx.x) >> 5);
    int lane = threadIdx.x & 31;
    if (wid >= e) return;
    int s = src[wid], d = dst[wid];
    float w = dinv[s] * dinv[d];
    const float4 hv = *(const float4*)&hW[(size_t)s * HIDW + 4 * lane];
    float* o = &out[(size_t)d * HIDW + 4 * lane];
    atomicAdd(o + 0, hv.x * w);
    atomicAdd(o + 1, hv.y * w);
    atomicAdd(o + 2, hv.z * w);
    atomicAdd(o + 3, hv.w * w);
}
__global__ void k_relu(float* __restrict__ x, long n) {
    long i = (long)blockIdx.x * blockDim.x + threadIdx.x;
    if (i < n) x[i] = fmaxf(x[i], 0.f);
}

// ---------------- GAT ----------------
__global__ void k_gat_scores(const float* __restrict__ hW, const float* __restrict__ aS,
                             const float* __restrict__ aD, float* __restrict__ alS,
                             float* __restrict__ alD, int n) {
    long i = (long)blockIdx.x * blockDim.x + threadIdx.x;
    if (i >= (long)n * NHEAD) return;
    int node = (int)(i >> 3), h = (int)(i & 7);
    const float* hv = &hW[(size_t)node * HIDW + h * CHAN];
    float ss = 0.f, sd = 0.f;
#pragma unroll
    for (int c = 0; c < CHAN; ++c) {
        float v = hv[c];
        ss += v * aS[h * CHAN + c];
        sd += v * aD[h * CHAN + c];
    }
    alS[i] = ss; alD[i] = sd;
}
__global__ void k_gat_init_m(const float* __restrict__ alS, const float* __restrict__ alD,
                             float* __restrict__ m, int n) {
    long i = (long)blockIdx.x * blockDim.x + threadIdx.x;
    if (i < (long)n * NHEAD) m[i] = leakyf(alS[i] + alD[i]);   // self-loop edge
}
__global__ void k_gat_edge_max(const int* __restrict__ src, const int* __restrict__ dst,
                               const float* __restrict__ alS, const float* __restrict__ alD,
                               float* __restrict__ m, int e) {
    long i = (long)blockIdx.x * blockDim.x + threadIdx.x;
    if (i >= (long)e * NHEAD) return;
    int ei = (int)(i >> 3), h = (int)(i & 7);
    int s = src[ei], d = dst[ei];
    float v = leakyf(alS[s * NHEAD + h] + alD[d * NHEAD + h]);
    atomicMaxF(&m[d * NHEAD + h], v);
}
__global__ void k_gat_init_den(const float* __restrict__ alS, const float* __restrict__ alD,
                               const float* __restrict__ m, float* __restrict__ den, int n) {
    long i = (long)blockIdx.x * blockDim.x + threadIdx.x;
    if (i < (long)n * NHEAD) den[i] = expf(leakyf(alS[i] + alD[i]) - m[i]);
}
__global__ void k_gat_edge_den(const int* __restrict__ src, const int* __restrict__ dst,
                               const float* __restrict__ alS, const float* __restrict__ alD,
                               const float* __restrict__ m, float* __restrict__ den, int e) {
    long i = (long)blockIdx.x * blockDim.x + threadIdx.x;
    if (i >= (long)e * NHEAD) return;
    int ei = (int)(i >> 3), h = (int)(i & 7);
    int s = src[ei], d = dst[ei];
    float v = leakyf(alS[s * NHEAD + h] + alD[d * NHEAD + h]);
    atomicAdd(&den[d * NHEAD + h], expf(v - m[d * NHEAD + h]));
}
__global__ void k_gat_init_out(const float* __restrict__ hW, const float* __restrict__ alS,
                               const float* __restrict__ alD, const float* __restrict__ m,
                               const float* __restrict__ den, float* __restrict__ out, int n) {
    long i = (long)blockIdx.x * blockDim.x + threadIdx.x;
    if (i >= (long)n * HIDW) return;
    int node = (int)(i >> 7), h = (int)((i >> 4) & 7);
    float ev   = leakyf(alS[node * NHEAD + h] + alD[node * NHEAD + h]);
    float coef = expf(ev - m[node * NHEAD + h]) / den[node * NHEAD + h];
    out[i] = hW[i] * coef;
}
__global__ __launch_bounds__(256)
void k_gat_scatter(const float* __restrict__ hW, const int* __restrict__ src,
                   const int* __restrict__ dst, const float* __restrict__ alS,
                   const float* __restrict__ alD, const float* __restrict__ m,
                   const float* __restrict__ den, float* __restrict__ out, int e) {
    int wid  = (int)(((long)blockIdx.x * blockDim.x + threadIdx.x) >> 5);
    int lane = threadIdx.x & 31;
    if (wid >= e) return;
    int s = src[wid], d = dst[wid];
    int head = lane >> 2;                      // 4 features per lane, 16 per head
    float ev   = leakyf(alS[s * NHEAD + head] + alD[d * NHEAD + head]);
    float coef = expf(ev - m[d * NHEAD + head]) / den[d * NHEAD + head];
    const float4 hv = *(const float4*)&hW[(size_t)s * HIDW + 4 * lane];
    float* o = &out[(size_t)d * HIDW + 4 * lane];
    atomicAdd(o + 0, hv.x * coef);
    atomicAdd(o + 1, hv.y * coef);
    atomicAdd(o + 2, hv.z * coef);
    atomicAdd(o + 3, hv.w * coef);
}
__global__ void k_gat_final(const float* __restrict__ out, const float* __restrict__ bias,
                            float* __restrict__ hnext, int n) {
    long i = (long)blockIdx.x * blockDim.x + threadIdx.x;
    if (i >= (long)n * CHAN) return;
    int node = (int)(i >> 4), c = (int)(i & 15);
    float sum = 0.f;
#pragma unroll
    for (int h = 0; h < NHEAD; ++h) sum += out[(size_t)node * HIDW + h * CHAN + c];
    hnext[i] = sum * (1.f / NHEAD) + bias[c];
}

// ---------------- output projection 16 -> 7 ----------------
__global__ void k_logits(const float* __restrict__ h, const float* __restrict__ W,
                         const float* __restrict__ b, float* __restrict__ out, int n) {
    int i = blockIdx.x * blockDim.x + threadIdx.x;
    if (i >= n) return;
    float hv[CHAN];
#pragma unroll
    for (int c = 0; c < CHAN; ++c) hv[c] = h[(size_t)i * CHAN + c];
#pragma unroll
    for (int j = 0; j < NCLS; ++j) {
        float acc = b[j];
#pragma unroll
        for (int c = 0; c < CHAN; ++c) acc += hv[c] * W[c * NCLS + j];
        out[(size_t)i * NCLS + j] = acc;
    }
}

static inline int cdiv(long a, long b) { return (int)((a + b - 1) / b); }
static inline int padK(int K) { return (K + 3) & ~3; }

extern "C" void kernel_launch(void* const* d_in, const int* in_sizes, int n_in,
                              void* d_out, int out_size, void* d_ws, size_t ws_size,
                              hipStream_t stream) {
    const float* x  = (const float*)d_in[0];
    const int*   ei = (const int*)d_in[1];
    const int E = in_sizes[1] / 2;
    const int n = in_sizes[0] / F_INN;
    const int* src = ei;
    const int* dst = ei + E;

    // scratch layout (~88 MB)
    float* ws   = (float*)d_ws;
    float* dinv = ws;  ws += n;
    float* bufA = ws;  ws += (size_t)n * HIDW;   // GEMM output
    float* bufB = ws;  ws += (size_t)n * HIDW;   // GCN aggregate
    float* bufC = ws;  ws += (size_t)n * HIDW;   // GAT aggregate
    float* hcur = ws;  ws += (size_t)n * CHAN;   // layer output (N x 16)
    float* alS  = ws;  ws += (size_t)n * NHEAD;
    float* alD  = ws;  ws += (size_t)n * NHEAD;
    float* mM   = ws;  ws += (size_t)n * NHEAD;
    float* den  = ws;  ws += (size_t)n * NHEAD;
    float* Bt   = ws;  ws += (size_t)HIDW * padK(F_INN);  // transposed weight staging

    // symmetric GCN normalization (deg includes self-loop)
    k_init_deg <<<cdiv(n, 256), 256, 0, stream>>>(dinv, n);
    k_deg_edges<<<cdiv(E, 256), 256, 0, stream>>>(dst, dinv, E);
    k_rsqrt    <<<cdiv(n, 256), 256, 0, stream>>>(dinv, n);

    const float* hin = x;
    int Kin = F_INN;
    for (int l = 0; l < 3; ++l) {
        const float* gcnW  = (const float*)d_in[2 + 6 * l + 0];
        const float* gcnB  = (const float*)d_in[2 + 6 * l + 1];
        const float* gatW  = (const float*)d_in[2 + 6 * l + 2];
        const float* gatAS = (const float*)d_in[2 + 6 * l + 3];
        const float* gatAD = (const float*)d_in[2 + 6 * l + 4];
        const float* gatB  = (const float*)d_in[2 + 6 * l + 5];

        // ---- GCN: bufB = relu(Dinv (A+I) Dinv (hin @ W) + b) ----
        int Kp = padK(Kin);
        k_transposeB  <<<cdiv((long)HIDW * Kp, 256), 256, 0, stream>>>(gcnW, Bt, Kin, Kp);
        k_gemm_wmma128<<<cdiv(n, 16), 256, 0, stream>>>(hin, Bt, bufA, n, Kin, Kp);
        k_gcn_init    <<<cdiv((long)n * HIDW, 256), 256, 0, stream>>>(bufA, dinv, gcnB, bufB, n);
        k_gcn_scatter <<<cdiv(E, 8), 256, 0, stream>>>(bufA, dinv, src, dst, bufB, E);
        k_relu        <<<cdiv((long)n * HIDW, 256), 256, 0, stream>>>(bufB, (long)n * HIDW);

        // ---- GAT: hcur = mean_h softmax-attention aggregate + b ----
        k_transposeB  <<<cdiv((long)HIDW * HIDW, 256), 256, 0, stream>>>(gatW, Bt, HIDW, HIDW);
        k_gemm_wmma128<<<cdiv(n, 16), 256, 0, stream>>>(bufB, Bt, bufA, n, HIDW, HIDW);
        k_gat_scores  <<<cdiv((long)n * NHEAD, 256), 256, 0, stream>>>(bufA, gatAS, gatAD, alS, alD, n);
        k_gat_init_m  <<<cdiv((long)n * NHEAD, 256), 256, 0, stream>>>(alS, alD, mM, n);
        k_gat_edge_max<<<cdiv((long)E * NHEAD, 256), 256, 0, stream>>>(src, dst, alS, alD, mM, E);
        k_gat_init_den<<<cdiv((long)n * NHEAD, 256), 256, 0, stream>>>(alS, alD, mM, den, n);
        k_gat_edge_den<<<cdiv((long)E * NHEAD, 256), 256, 0, stream>>>(src, dst, alS, alD, mM, den, E);
        k_gat_init_out<<<cdiv((long)n * HIDW, 256), 256, 0, stream>>>(bufA, alS, alD, mM, den, bufC, n);
        k_gat_scatter <<<cdiv(E, 8), 256, 0, stream>>>(bufA, src, dst, alS, alD, mM, den, bufC, E);
        k_gat_final   <<<cdiv((long)n * CHAN, 256), 256, 0, stream>>>(bufC, gatB, hcur, n);

        hin = hcur;
        Kin = CHAN;
    }

    const float* outW = (const float*)d_in[20];
    const float* outB = (const float*)d_in[21];
    k_logits<<<cdiv(n, 256), 256, 0, stream>>>(hcur, outW, outB, (float*)d_out, n);
}